// TemporalDecoderLSTM_16346645529090
// MI455X (gfx1250) — compile-verified
//
#include <hip/hip_runtime.h>

#define B_AG 32768
#define B2   65536          // B * 2
#define OBS  20
#define PRED 30

typedef float v2f __attribute__((ext_vector_type(2)));
typedef float v8f __attribute__((ext_vector_type(8)));

// ---------------------------------------------------------------------------
// Pack weights.
//  [0,4096)      : W_ih^T B-operand tiles  packed_ih[nt][kk][lane][v]
//  [4096,20480)  : W_hh^T B-operand tiles  packed_hh[nt][kk][lane][v]
//  bsum[256]     : b_ih + b_hh
//  pemb[640]     : emb_W packed per (j, khalf): pemb[(j*2+kh)*8 + q*2+p]
//                   = emb_W[(4q + 2kh + p)*40 + j]
//  pembb[16]     : emb_b packed: pembb[kh*8 + q*2+p] = emb_b[4q+2kh+p]
// B-tile layout for V_WMMA_F32_16X16X4_F32 (4x16, K rows x N cols):
//   VGPR0: lanes 0-15 -> K=k0+0, lanes 16-31 -> K=k0+2; VGPR1: +1 each.
// ---------------------------------------------------------------------------
__global__ __launch_bounds__(256) void pack_kernel(
    const float* __restrict__ W_ih, const float* __restrict__ W_hh,
    const float* __restrict__ b_ih, const float* __restrict__ b_hh,
    const float* __restrict__ emb_W, const float* __restrict__ emb_b,
    float* __restrict__ wpack, float* __restrict__ bsum,
    float* __restrict__ pemb, float* __restrict__ pembb)
{
  int idx = blockIdx.x * 256 + threadIdx.x;
  if (idx < 4096) {                       // W_ih part: 16 ntiles x 4 ksteps
    int nt = idx >> 8, kk = (idx >> 6) & 3, lane = (idx >> 1) & 31, v = idx & 1;
    int k = 4 * kk + v + 2 * (lane >> 4);
    int n = nt * 16 + (lane & 15);
    wpack[idx] = W_ih[n * 16 + k];        // B[k][n] = W_ih[n][k]
  } else if (idx < 20480) {               // W_hh part: 16 ntiles x 16 ksteps
    int rel = idx - 4096;
    int nt = rel >> 10, kk = (rel >> 6) & 15, lane = (rel >> 1) & 31, v = rel & 1;
    int k = 4 * kk + v + 2 * (lane >> 4);
    int n = nt * 16 + (lane & 15);
    wpack[idx] = W_hh[n * 64 + k];        // B[k][n] = W_hh[n][k]
  } else if (idx < 20736) {
    int n = idx - 20480;
    bsum[n] = b_ih[n] + b_hh[n];
  } else if (idx < 21376) {               // packed emb_W
    int rel = idx - 20736;
    int j = rel >> 4, kh = (rel >> 3) & 1, q = (rel >> 1) & 3, p = rel & 1;
    pemb[rel] = emb_W[(4 * q + 2 * kh + p) * 40 + j];
  } else if (idx < 21392) {               // packed emb_b
    int rel = idx - 21376;
    int kh = rel >> 3, q = (rel >> 1) & 3, p = rel & 1;
    pembb[rel] = emb_b[4 * q + 2 * kh + p];
  }
}

// ---------------------------------------------------------------------------
// Init: circular traj buffer slabs 0..19 = traj_abs, h = decoder_h, c = 0.
// ---------------------------------------------------------------------------
__global__ __launch_bounds__(256) void init_kernel(
    const float* __restrict__ traj_abs, const float* __restrict__ dec_h,
    float* __restrict__ ws_traj, float* __restrict__ ws_h, float* __restrict__ ws_c)
{
  int idx = blockIdx.x * 256 + threadIdx.x;      // grid covers B*64 = 2097152
  if (idx < OBS * B2) ws_traj[idx] = traj_abs[idx];
  if (idx < B_AG * 64) { ws_h[idx] = dec_h[idx]; ws_c[idx] = 0.f; }
}

// ---------------------------------------------------------------------------
// Fused step: async-LDS weight staging overlapped with embed (LN40 + 40x16)
// -> fp32-WMMA LSTM gates -> cell math -> pos head (LN64 + 64x2 via 16-lane
// shuffle reductions) -> traj roll + d_out.
// 8 waves/block, 16 agents/wave (2 lanes/agent: mrow = lane&15, khalf = lane>>4).
// ---------------------------------------------------------------------------
__global__ __launch_bounds__(256) void step_kernel(
    float* traj,                          // circular 50-slab buffer (r+w)
    float* __restrict__ h, float* __restrict__ c,
    const float* __restrict__ wpack, const float* __restrict__ bsum,
    const float* __restrict__ pemb, const float* __restrict__ pembb,
    const float* __restrict__ g1, const float* __restrict__ b1,
    const float* __restrict__ h2pW, const float* __restrict__ h2pB,
    const float* __restrict__ g2, const float* __restrict__ b2,
    float* __restrict__ out, int t)
{
  __shared__ __align__(16) float lw[20480];     // packed W_ih | W_hh  (80 KB)
  __shared__ float lb[256];                     // b_ih + b_hh
  __shared__ __align__(16) float lpe[640];      // packed emb_W
  __shared__ float lpeb[16];                    // packed emb_b
  __shared__ float lg1[40], lb1[40];
  __shared__ float lh2w[128], lg2[64], lb2[64], lh2b[2];

  // ---- async DMA of the 80KB packed gate weights into LDS (ASYNCcnt) ------
  {
    unsigned lbase = (unsigned)(uintptr_t)&lw[0];
    unsigned long long gbase = (unsigned long long)(uintptr_t)wpack;
    #pragma unroll
    for (int i = 0; i < 20; ++i) {
      unsigned off = ((unsigned)threadIdx.x + i * 256u) * 16u;
      unsigned laddr = lbase + off;
      asm volatile("global_load_async_to_lds_b128 %0, %1, %2 offset:0"
                   :: "v"(laddr), "v"(off), "s"(gbase) : "memory");
    }
  }
  // ---- small params synchronously (needed by the embed phase) -------------
  {
    lb[threadIdx.x] = bsum[threadIdx.x];
    const float4* p4 = (const float4*)pemb;
    float4* e4 = (float4*)lpe;
    if (threadIdx.x < 160) e4[threadIdx.x] = p4[threadIdx.x];
    if (threadIdx.x < 16)  lpeb[threadIdx.x] = pembb[threadIdx.x];
    if (threadIdx.x < 40)  { lg1[threadIdx.x] = g1[threadIdx.x];
                             lb1[threadIdx.x] = b1[threadIdx.x]; }
    if (threadIdx.x < 128) lh2w[threadIdx.x] = h2pW[threadIdx.x];
    if (threadIdx.x < 64)  { lg2[threadIdx.x] = g2[threadIdx.x];
                             lb2[threadIdx.x] = b2[threadIdx.x]; }
    if (threadIdx.x < 2)   lh2b[threadIdx.x] = h2pB[threadIdx.x];
  }
  __syncthreads();   // small arrays visible; big async fill still in flight

  const int lane  = threadIdx.x & 31;
  const int wave  = threadIdx.x >> 5;
  const int tile  = blockIdx.x * 8 + wave;
  const int abase = tile * 16;
  const int mrow  = lane & 15;
  const int khalf = lane >> 4;

  // ---------------- embed: produce this lane's A_x slices directly ---------
  // (overlaps with the async weight DMA)
  v2f ax[4];
  {
    const float4* w4 = (const float4*)(traj + (size_t)t * B2 +
                                       (size_t)(abase + mrow) * 40);
    float wv[40];
    float s = 0.f;
    #pragma unroll
    for (int i = 0; i < 10; ++i) {
      float4 q = w4[i];
      wv[4*i+0] = q.x; wv[4*i+1] = q.y; wv[4*i+2] = q.z; wv[4*i+3] = q.w;
      s += q.x + q.y + q.z + q.w;
    }
    float m = s * (1.f / 40.f);
    float ss = 0.f;
    #pragma unroll
    for (int j = 0; j < 40; ++j) { float d = wv[j] - m; ss += d * d; }
    float istd = rsqrtf(ss * (1.f / 40.f) + 1e-5f);

    float ex[8];
    #pragma unroll
    for (int i = 0; i < 8; ++i) ex[i] = lpeb[khalf * 8 + i];
    for (int j = 0; j < 40; ++j) {
      float ln = (wv[j] - m) * istd * lg1[j] + lb1[j];
      const float* pw = lpe + (j * 2 + khalf) * 8;
      #pragma unroll
      for (int i = 0; i < 8; ++i) ex[i] = fmaf(ln, pw[i], ex[i]);
    }
    #pragma unroll
    for (int i = 0; i < 8; ++i) ex[i] = ex[i] > 0.f ? ex[i] : 0.01f * ex[i];
    #pragma unroll
    for (int q = 0; q < 4; ++q) ax[q] = (v2f){ex[2 * q], ex[2 * q + 1]};
  }

  // ---------------- A_h slices from previous h ([B][64] agent-major) -------
  v2f ah[16];
  {
    const float* hb = h + (size_t)(abase + mrow) * 64 + 2 * khalf;
    #pragma unroll
    for (int kk = 0; kk < 16; ++kk) ah[kk] = *(const v2f*)(hb + 4 * kk);
  }

  // ---- wait for async weight DMA, then barrier so all fills are visible ---
  asm volatile("s_wait_asynccnt 0x0" ::: "memory");
  __syncthreads();

  // ---------------- WMMA gates + cell math ---------------------------------
  float hreg[4][8];                      // h tile values, kept for pos head
  for (int j = 0; j < 4; ++j) {          // 16-wide column chunk of H
    v8f acc[4];                          // i, f, g, o tiles
    #pragma unroll
    for (int g = 0; g < 4; ++g) {
      int nt = g * 4 + j;                // i: nt 0..3, f: 4..7, g: 8..11, o: 12..15
      float bv = lb[nt * 16 + mrow];
      acc[g] = (v8f){bv, bv, bv, bv, bv, bv, bv, bv};
      const float* wi = lw + nt * 256 + lane * 2;
      #pragma unroll
      for (int kk = 0; kk < 4; ++kk) {   // x @ W_ih^T  (K=16)
        v2f bb = *(const v2f*)(wi + kk * 64);
        acc[g] = __builtin_amdgcn_wmma_f32_16x16x4_f32(
            false, ax[kk], false, bb, (short)0, acc[g], false, false);
      }
      const float* wh = lw + 4096 + nt * 1024 + lane * 2;
      #pragma unroll
      for (int kk = 0; kk < 16; ++kk) {  // h @ W_hh^T  (K=64)
        v2f bb = *(const v2f*)(wh + kk * 64);
        acc[g] = __builtin_amdgcn_wmma_f32_16x16x4_f32(
            false, ah[kk], false, bb, (short)0, acc[g], false, false);
      }
    }
    float* cb = c + (size_t)(tile * 4 + j) * 8 * 32 + lane;
    #pragma unroll
    for (int r = 0; r < 8; ++r) {
      float iv = acc[0][r], fv = acc[1][r], gv = acc[2][r], ov = acc[3][r];
      float c_old = cb[r * 32];
      float is = 1.f / (1.f + __expf(-iv));
      float fs = 1.f / (1.f + __expf(-fv));
      float os = 1.f / (1.f + __expf(-ov));
      float gt = tanhf(gv);
      float cn = fmaf(fs, c_old, is * gt);
      float hn = os * tanhf(cn);
      cb[r * 32] = cn;
      hreg[j][r] = hn;
      h[(size_t)(abase + r + 8 * khalf) * 64 + 16 * j + mrow] = hn;
    }
  }

  // ---------------- pos head: LN64 + 64->2, shuffle reductions -------------
  float w0v[4], w1v[4], g2v[4], b2v[4];
  #pragma unroll
  for (int j = 0; j < 4; ++j) {
    int n = 16 * j + mrow;
    w0v[j] = lh2w[n]; w1v[j] = lh2w[64 + n];
    g2v[j] = lg2[n];  b2v[j] = lb2[n];
  }
  float hb0 = lh2b[0], hb1 = lh2b[1];
  #pragma unroll
  for (int r = 0; r < 8; ++r) {
    float s = hreg[0][r] + hreg[1][r] + hreg[2][r] + hreg[3][r];
    s += __shfl_xor(s, 1, 16); s += __shfl_xor(s, 2, 16);
    s += __shfl_xor(s, 4, 16); s += __shfl_xor(s, 8, 16);
    float m = s * (1.f / 64.f);
    float ss = 0.f;
    #pragma unroll
    for (int j = 0; j < 4; ++j) { float d = hreg[j][r] - m; ss += d * d; }
    ss += __shfl_xor(ss, 1, 16); ss += __shfl_xor(ss, 2, 16);
    ss += __shfl_xor(ss, 4, 16); ss += __shfl_xor(ss, 8, 16);
    float istd = rsqrtf(ss * (1.f / 64.f) + 1e-5f);
    float p0 = 0.f, p1 = 0.f;
    #pragma unroll
    for (int j = 0; j < 4; ++j) {
      float ln = (hreg[j][r] - m) * istd * g2v[j] + b2v[j];
      p0 = fmaf(ln, w0v[j], p0);
      p1 = fmaf(ln, w1v[j], p1);
    }
    p0 += __shfl_xor(p0, 1, 16); p0 += __shfl_xor(p0, 2, 16);
    p0 += __shfl_xor(p0, 4, 16); p0 += __shfl_xor(p0, 8, 16);
    p1 += __shfl_xor(p1, 1, 16); p1 += __shfl_xor(p1, 2, 16);
    p1 += __shfl_xor(p1, 4, 16); p1 += __shfl_xor(p1, 8, 16);
    if (mrow == r) {
      int ag = abase + r + 8 * khalf;
      float2 p = make_float2(p0 + hb0, p1 + hb1);
      *(float2*)(traj + (size_t)(OBS + t) * B2 + (size_t)ag * 2) = p;
      *(float2*)(out + (size_t)t * B2 + (size_t)ag * 2) = p;
    }
  }
}

// ---------------------------------------------------------------------------
extern "C" void kernel_launch(void* const* d_in, const int* in_sizes, int n_in,
                              void* d_out, int out_size, void* d_ws, size_t ws_size,
                              hipStream_t stream)
{
  (void)in_sizes; (void)n_in; (void)out_size; (void)ws_size;
  const float* traj_abs  = (const float*)d_in[0];
  // d_in[1] = traj_rel (unused in this branch)
  const float* decoder_h = (const float*)d_in[2];
  const float* W_ih      = (const float*)d_in[3];
  const float* W_hh      = (const float*)d_in[4];
  const float* b_ih      = (const float*)d_in[5];
  const float* b_hh      = (const float*)d_in[6];
  const float* emb_W     = (const float*)d_in[7];
  const float* emb_b     = (const float*)d_in[8];
  const float* h2p_W     = (const float*)d_in[9];
  const float* h2p_b     = (const float*)d_in[10];
  const float* ln1_g     = (const float*)d_in[11];
  const float* ln1_b     = (const float*)d_in[12];
  const float* ln2_g     = (const float*)d_in[13];
  const float* ln2_b     = (const float*)d_in[14];
  float* out = (float*)d_out;

  float* ws      = (float*)d_ws;
  float* ws_traj = ws;                                   // 50*B2 = 3,276,800
  float* ws_h    = ws_traj + (size_t)(OBS + PRED) * B2;  // B*64  = 2,097,152
  float* ws_c    = ws_h + (size_t)B_AG * 64;             // B*64 (tile layout)
  float* ws_wp   = ws_c + (size_t)B_AG * 64;             // 20480
  float* ws_bs   = ws_wp + 20480;                        // 256
  float* ws_pe   = ws_bs + 256;                          // 640
  float* ws_peb  = ws_pe + 640;                          // 16

  pack_kernel<<<84, 256, 0, stream>>>(W_ih, W_hh, b_ih, b_hh, emb_W, emb_b,
                                      ws_wp, ws_bs, ws_pe, ws_peb);
  init_kernel<<<8192, 256, 0, stream>>>(traj_abs, decoder_h, ws_traj, ws_h, ws_c);

  for (int t = 0; t < PRED; ++t) {
    step_kernel<<<256, 256, 0, stream>>>(ws_traj, ws_h, ws_c, ws_wp, ws_bs,
                                         ws_pe, ws_peb, ln1_g, ln1_b,
                                         h2p_W, h2p_b, ln2_g, ln2_b, out, t);
  }
}